// DeepSeekV2MoE_63702954934743
// MI455X (gfx1250) — compile-verified
//
#include <hip/hip_runtime.h>
#include <hip/hip_bf16.h>
#include <math.h>

typedef float v2f __attribute__((ext_vector_type(2)));
typedef float v8f __attribute__((ext_vector_type(8)));

#define NE 16      // experts
#define NK 4       // top-k
#define DH 2048    // hidden
#define DI 1024    // intermediate
#define NT 2048    // tokens (B*S)
#define BM 32
#define BN 32
#define BK 32
#define LDP 33     // padded LDS stride (kills bank conflicts)

// ---- CDNA5 async global->LDS path (guarded: falls back to reg pipeline) ----
#if defined(__has_builtin)
#  if __has_builtin(__builtin_amdgcn_global_load_async_to_lds_b32)
#    define USE_ASYNC_LDS 1
#  endif
#endif
#ifndef USE_ASYNC_LDS
#  define USE_ASYNC_LDS 0
#endif

#if USE_ASYNC_LDS
#  if __has_builtin(__builtin_amdgcn_s_wait_asynccnt)
#    define WAIT_ASYNC(n) __builtin_amdgcn_s_wait_asynccnt(n)
#  else
#    define WAIT_ASYNC(n) asm volatile("s_wait_asynccnt %0" ::"i"(n) : "memory")
#  endif
typedef __attribute__((address_space(1))) int* gas_i32p;
typedef __attribute__((address_space(3))) int* las_i32p;
#  define GASYNC(gsrc, ldst)                                          \
     __builtin_amdgcn_global_load_async_to_lds_b32(                   \
         (gas_i32p)(gsrc), (las_i32p)(ldst), 0, 0)
#endif

// ---------------- Router: one wave per token ----------------
__global__ __launch_bounds__(32) void moe_router(
    const float* __restrict__ x, const float* __restrict__ gate_w,
    int* __restrict__ topk_ids, float* __restrict__ topk_w,
    int* __restrict__ counts) {
  const int t = blockIdx.x;
  const int lane = threadIdx.x;
  const float* xr = x + (size_t)t * DH;

  float acc[NE];
#pragma unroll
  for (int e = 0; e < NE; ++e) acc[e] = 0.f;

  for (int h = lane; h < DH; h += 32) {
    float xv = xr[h];
#pragma unroll
    for (int e = 0; e < NE; ++e) acc[e] = fmaf(xv, gate_w[e * DH + h], acc[e]);
  }
#pragma unroll
  for (int e = 0; e < NE; ++e) {
    float v = acc[e];
#pragma unroll
    for (int off = 16; off > 0; off >>= 1) v += __shfl_xor(v, off, 32);
    acc[e] = v;
  }
  if (lane == 0) {
    int ids[NK]; float vals[NK];
    unsigned used = 0;
#pragma unroll
    for (int k = 0; k < NK; ++k) {
      float best = -INFINITY; int bi = 0;
      for (int e = 0; e < NE; ++e)
        if (!((used >> e) & 1u) && acc[e] > best) { best = acc[e]; bi = e; }
      used |= (1u << bi); ids[k] = bi; vals[k] = best;
    }
    float s = 0.f, w[NK];
#pragma unroll
    for (int k = 0; k < NK; ++k) { w[k] = __expf(vals[k] - vals[0]); s += w[k]; }
    float inv = 1.f / s;
#pragma unroll
    for (int k = 0; k < NK; ++k) {
      topk_ids[t * NK + k] = ids[k];
      topk_w[t * NK + k]   = w[k] * inv;
      atomicAdd(&counts[ids[k]], 1);
    }
  }
}

// ---------------- Prefix sum over 16 expert counts ----------------
__global__ void moe_offsets(const int* __restrict__ counts, int* __restrict__ offsets) {
  if (threadIdx.x == 0 && blockIdx.x == 0) {
    int s = 0;
    for (int e = 0; e < NE; ++e) { offsets[e] = s; s += counts[e]; }
    offsets[NE] = s;
  }
}

// ---- Deterministic grouped-token-list build: one wave per expert ----
__global__ __launch_bounds__(32) void moe_build(
    const int* __restrict__ topk_ids, const float* __restrict__ topk_w,
    const int* __restrict__ offsets,
    int* __restrict__ perm, float* __restrict__ pgate) {
  const int e = blockIdx.x;
  const int lane = threadIdx.x;
  const int base = offsets[e];
  int running = 0;
  for (int t0 = 0; t0 < NT; t0 += 32) {
    int t = t0 + lane;
    bool sel = false; float wk = 0.f;
#pragma unroll
    for (int k = 0; k < NK; ++k)
      if (topk_ids[t * NK + k] == e) { sel = true; wk = topk_w[t * NK + k]; }
    unsigned m = (unsigned)__ballot(sel);       // wave32: low 32 bits
    int pfx = __popc(m & ((1u << lane) - 1u));
    if (sel) { perm[base + running + pfx] = t; pgate[base + running + pfx] = wk; }
    running += __popc(m);
  }
}

// -------- GEMM1: h[s, i] = silu(x@w1g^T) * (x@w1u^T) * gate, per expert --------
__global__ __launch_bounds__(128) void moe_gemm1(
    const float* __restrict__ x, const float* __restrict__ w1g,
    const float* __restrict__ w1u, const int* __restrict__ counts,
    const int* __restrict__ offsets, const int* __restrict__ perm,
    const float* __restrict__ pgate, float* __restrict__ hbuf) {
  const int e   = blockIdx.z;
  const int cnt = counts[e];
  const int mt  = blockIdx.y;
  if (mt * BM >= cnt) return;                    // uniform early exit
  const int i0   = blockIdx.x * BN;
  const int base = offsets[e];

  const int tid  = threadIdx.x;
  const int lane = tid & 31;
  const int wave = tid >> 5;
  const int wm = wave >> 1, wn = wave & 1;
  const int half = lane >> 4, l15 = lane & 15;
  const int kcol = tid & 31;                     // contiguous global loads
  const int rbase = tid >> 5;                    // 4 rows per pass, 8 passes

  // Hoist per-row gather offsets; invalid rows clamp to a valid slot (their
  // finite junk output is masked in the epilogue -> no mask multiply needed).
  size_t aOff[8];
#pragma unroll
  for (int j = 0; j < 8; ++j) {
    int s = mt * BM + rbase + j * 4;
    int sc = (s < cnt) ? s : (cnt - 1);
    aOff[j] = (size_t)perm[base + sc] * DH;
  }

  const float* w1ge = w1g + (size_t)e * DI * DH;
  const float* w1ue = w1u + (size_t)e * DI * DH;

  v8f cg = {}; v8f cu = {};

#if USE_ASYNC_LDS
  __shared__ float As[2][BM][LDP];
  __shared__ float Bg[2][BM][LDP];
  __shared__ float Bu[2][BM][LDP];

#define ISSUE1(BUF, H0)                                             \
  _Pragma("unroll")                                                 \
  for (int j = 0; j < 8; ++j) {                                     \
    int r = rbase + j * 4;                                          \
    GASYNC(x + aOff[j] + (H0) + kcol, &As[BUF][r][kcol]);           \
    size_t off = (size_t)(i0 + r) * DH + (H0) + kcol;               \
    GASYNC(w1ge + off, &Bg[BUF][r][kcol]);                          \
    GASYNC(w1ue + off, &Bu[BUF][r][kcol]);                          \
  }

  ISSUE1(0, 0);
  for (int h0 = 0; h0 < DH; h0 += BK) {
    const int buf = (h0 / BK) & 1;
    if (h0 + BK < DH) { ISSUE1(buf ^ 1, h0 + BK); WAIT_ASYNC(24); }
    else              { WAIT_ASYNC(0); }
    __syncthreads();                              // all waves' tile landed
    const float (*pA)[LDP] = As[buf];
    const float (*pG)[LDP] = Bg[buf];
    const float (*pU)[LDP] = Bu[buf];
#else
  __shared__ float As[BM][LDP];
  __shared__ float Bg[BN][LDP];
  __shared__ float Bu[BN][LDP];

  float aR[8], gR[8], uR[8];
#define LOAD_TILE1(H0)                                              \
  _Pragma("unroll")                                                 \
  for (int j = 0; j < 8; ++j) {                                     \
    int r = rbase + j * 4;                                          \
    aR[j] = x[aOff[j] + (H0) + kcol];                               \
    size_t off = (size_t)(i0 + r) * DH + (H0) + kcol;               \
    gR[j] = w1ge[off];                                              \
    uR[j] = w1ue[off];                                              \
    __builtin_prefetch(&w1ge[off + 4 * BK], 0, 0);                  \
    __builtin_prefetch(&w1ue[off + 4 * BK], 0, 0);                  \
  }

  LOAD_TILE1(0);
  for (int h0 = 0; h0 < DH; h0 += BK) {
#pragma unroll
    for (int j = 0; j < 8; ++j) {
      int r = rbase + j * 4;
      As[r][kcol] = aR[j];
      Bg[r][kcol] = gR[j];
      Bu[r][kcol] = uR[j];
    }
    __syncthreads();
    if (h0 + BK < DH) { LOAD_TILE1(h0 + BK); }   // overlap with WMMA below
    const float (*pA)[LDP] = As;
    const float (*pG)[LDP] = Bg;
    const float (*pU)[LDP] = Bu;
#endif
#pragma unroll
    for (int kk = 0; kk < BK; kk += 4) {
      v2f a, bg, bu;
      int ar = wm * 16 + l15;
      a.x  = pA[ar][kk + 2 * half];
      a.y  = pA[ar][kk + 2 * half + 1];
      int bc = wn * 16 + l15;
      bg.x = pG[bc][kk + 2 * half];
      bg.y = pG[bc][kk + 2 * half + 1];
      bu.x = pU[bc][kk + 2 * half];
      bu.y = pU[bc][kk + 2 * half + 1];
      cg = __builtin_amdgcn_wmma_f32_16x16x4_f32(false, a, false, bg, (short)0, cg, false, false);
      cu = __builtin_amdgcn_wmma_f32_16x16x4_f32(false, a, false, bu, (short)0, cu, false, false);
    }
    __syncthreads();
  }
  // Epilogue: SwiGLU + fold gate weight into h (gate is linear w.r.t. down-proj)
#pragma unroll
  for (int r = 0; r < 8; ++r) {
    int row = wm * 16 + r + 8 * half;
    int s = mt * BM + row;
    if (s < cnt) {
      float g = cg[r], u = cu[r];
      float hv = (g / (1.f + __expf(-g))) * u * pgate[base + s];
      hbuf[(size_t)(base + s) * DI + i0 + wn * 16 + l15] = hv;
    }
  }
}

// -------- GEMM2: out[tok, :] += h[s, :] @ w2[e]^T (scatter-add) --------
__global__ __launch_bounds__(128) void moe_gemm2(
    const float* __restrict__ hbuf, const float* __restrict__ w2,
    const int* __restrict__ counts, const int* __restrict__ offsets,
    const int* __restrict__ perm, float* __restrict__ out) {
  const int e   = blockIdx.z;
  const int cnt = counts[e];
  const int mt  = blockIdx.y;
  if (mt * BM >= cnt) return;
  const int n0   = blockIdx.x * BN;              // output-hidden tile
  const int base = offsets[e];

  const int tid  = threadIdx.x;
  const int lane = tid & 31;
  const int wave = tid >> 5;
  const int wm = wave >> 1, wn = wave & 1;
  const int half = lane >> 4, l15 = lane & 15;
  const int kcol = tid & 31;
  const int rbase = tid >> 5;

  size_t aRow[8];
#pragma unroll
  for (int j = 0; j < 8; ++j) {
    int s = mt * BM + rbase + j * 4;
    int sc = (s < cnt) ? s : (cnt - 1);
    aRow[j] = (size_t)(base + sc) * DI;
  }

  const float* w2e = w2 + (size_t)e * DH * DI;
  v8f c = {};

#if USE_ASYNC_LDS
  __shared__ float As[2][BM][LDP];
  __shared__ float Bs[2][BM][LDP];

#define ISSUE2(BUF, K0)                                             \
  _Pragma("unroll")                                                 \
  for (int j = 0; j < 8; ++j) {                                     \
    int r = rbase + j * 4;                                          \
    GASYNC(hbuf + aRow[j] + (K0) + kcol, &As[BUF][r][kcol]);        \
    size_t off = (size_t)(n0 + r) * DI + (K0) + kcol;               \
    GASYNC(w2e + off, &Bs[BUF][r][kcol]);                           \
  }

  ISSUE2(0, 0);
  for (int k0 = 0; k0 < DI; k0 += BK) {
    const int buf = (k0 / BK) & 1;
    if (k0 + BK < DI) { ISSUE2(buf ^ 1, k0 + BK); WAIT_ASYNC(16); }
    else              { WAIT_ASYNC(0); }
    __syncthreads();
    const float (*pA)[LDP] = As[buf];
    const float (*pB)[LDP] = Bs[buf];
#else
  __shared__ float As[BM][LDP];
  __shared__ float Bs[BN][LDP];

  float aR[8], bR[8];
#define LOAD_TILE2(K0)                                              \
  _Pragma("unroll")                                                 \
  for (int j = 0; j < 8; ++j) {                                     \
    int r = rbase + j * 4;                                          \
    aR[j] = hbuf[aRow[j] + (K0) + kcol];                            \
    size_t off = (size_t)(n0 + r) * DI + (K0) + kcol;               \
    bR[j] = w2e[off];                                               \
    __builtin_prefetch(&w2e[off + 4 * BK], 0, 0);                   \
  }

  LOAD_TILE2(0);
  for (int k0 = 0; k0 < DI; k0 += BK) {
#pragma unroll
    for (int j = 0; j < 8; ++j) {
      int r = rbase + j * 4;
      As[r][kcol] = aR[j];
      Bs[r][kcol] = bR[j];
    }
    __syncthreads();
    if (k0 + BK < DI) { LOAD_TILE2(k0 + BK); }   // overlap with WMMA below
    const float (*pA)[LDP] = As;
    const float (*pB)[LDP] = Bs;
#endif
#pragma unroll
    for (int kk = 0; kk < BK; kk += 4) {
      v2f a, b;
      int ar = wm * 16 + l15;
      a.x = pA[ar][kk + 2 * half];
      a.y = pA[ar][kk + 2 * half + 1];
      int bc = wn * 16 + l15;
      b.x = pB[bc][kk + 2 * half];
      b.y = pB[bc][kk + 2 * half + 1];
      c = __builtin_amdgcn_wmma_f32_16x16x4_f32(false, a, false, b, (short)0, c, false, false);
    }
    __syncthreads();
  }
#pragma unroll
  for (int r = 0; r < 8; ++r) {
    int row = wm * 16 + r + 8 * half;
    int s = mt * BM + row;
    if (s < cnt) {
      int tok = perm[base + s];
      float* p = &out[(size_t)tok * DH + n0 + wn * 16 + l15];
      __hip_atomic_fetch_add(p, c[r], __ATOMIC_RELAXED, __HIP_MEMORY_SCOPE_AGENT);
    }
  }
}

extern "C" void kernel_launch(void* const* d_in, const int* in_sizes, int n_in,
                              void* d_out, int out_size, void* d_ws, size_t ws_size,
                              hipStream_t stream) {
  (void)in_sizes; (void)n_in; (void)ws_size;
  const float* x       = (const float*)d_in[0];
  const float* gate_w  = (const float*)d_in[1];
  const float* w1_gate = (const float*)d_in[2];
  const float* w1_up   = (const float*)d_in[3];
  const float* w2      = (const float*)d_in[4];
  float* out = (float*)d_out;

  // Workspace layout (all 4-byte types; total ~34 MB)
  int*   counts   = (int*)d_ws;                       // 16
  int*   offsets  = counts + 16;                      // 17 (pad to 32)
  int*   topk_ids = offsets + 32;                     // NT*NK
  float* topk_w   = (float*)(topk_ids + NT * NK);     // NT*NK
  int*   perm     = (int*)(topk_w + NT * NK);         // NT*NK
  float* pgate    = (float*)(perm + NT * NK);         // NT*NK
  float* hbuf     = pgate + NT * NK;                  // NT*NK*DI floats

  hipMemsetAsync(counts, 0, 64 * sizeof(int), stream);
  hipMemsetAsync(d_out, 0, (size_t)out_size * sizeof(float), stream);

  moe_router<<<NT, 32, 0, stream>>>(x, gate_w, topk_ids, topk_w, counts);
  moe_offsets<<<1, 32, 0, stream>>>(counts, offsets);
  moe_build<<<NE, 32, 0, stream>>>(topk_ids, topk_w, offsets, perm, pgate);

  dim3 g1(DI / BN, NT / BM, NE);
  moe_gemm1<<<g1, 128, 0, stream>>>(x, w1_gate, w1_up, counts, offsets, perm, pgate, hbuf);

  dim3 g2(DH / BN, NT / BM, NE);
  moe_gemm2<<<g2, 128, 0, stream>>>(hbuf, w2, counts, offsets, perm, out);
}